// FieldAwareFactorizationMachineLayer_53437983097345
// MI455X (gfx1250) — compile-verified
//
#include <hip/hip_runtime.h>

// FFM pairwise cross-interaction, MI455X (gfx1250, wave32).
// out[b, p(i,j), d] = W[j, x[b,i]+off[i], d] * W[i, x[b,j]+off[j], d]
// Store-bandwidth bound: 265 MB NT b128 stores (~11.4 us floor @ 23.3 TB/s);
// gather reads hit L2 (hot set = 23 tables * 23 field ranges * 100 rows * 128B ~= 6.8 MB << 192 MB L2).

#define NF   23
#define NP   253          // NF*(NF-1)/2
#define ED   32
#define VOC  190001
#define NB   8192
#define TPB  256

typedef float v4f __attribute__((ext_vector_type(4)));

// Compile-time pair table: p -> (i<<8)|j for i<j, i-major (matches jnp.triu_indices order).
struct PairTab {
    unsigned short v[NP];
    constexpr PairTab() : v() {
        int p = 0;
        for (int i = 0; i < NF; ++i)
            for (int j = i + 1; j < NF; ++j)
                v[p++] = (unsigned short)((i << 8) | j);
    }
};
__constant__ PairTab kPairs;   // zero-arg constexpr ctor runs at compile time

__global__ __launch_bounds__(TPB)
void ffm_cross_kernel(const int* __restrict__ x,
                      const int* __restrict__ offsets,
                      const float* __restrict__ W,
                      float* __restrict__ out)
{
    __shared__ int s_x[NF];
    __shared__ int s_adj[NF];
    __shared__ unsigned short s_pair[NP];

    const int tid = threadIdx.x;
    const int b   = blockIdx.x;

    // --- Stage this row's 23 raw indices into LDS via the CDNA5 async path ---
    if (tid < NF) {
        unsigned           lds_addr = (unsigned)(size_t)(&s_x[tid]);   // low 32b of generic ptr = LDS offset
        unsigned long long gaddr    = (unsigned long long)(size_t)(x + b * NF + tid);
        asm volatile("global_load_async_to_lds_b32 %0, %1, off"
                     :: "v"(lds_addr), "v"(gaddr)
                     : "memory");
    }

    // --- Copy constant pair table into LDS (broadcast-friendly ds_load_u16 in hot loop) ---
    if (tid < NP) s_pair[tid] = kPairs.v[tid];

    asm volatile("s_wait_asynccnt 0" ::: "memory");   // drain ASYNCcnt (only issuing wave stalls)
    __syncthreads();

    if (tid < NF) s_adj[tid] = s_x[tid] + offsets[tid];
    __syncthreads();

    const v4f* __restrict__ Wv = (const v4f*)W;
    v4f* __restrict__ outv = (v4f*)out + (size_t)b * (NP * (ED / 4));

    // 2024 work items per row: work = pair*8 + quad; 8 consecutive lanes cover one
    // 128B cacheline per gather operand and one 128B NT store.
    #pragma unroll
    for (int k = 0; k < 8; ++k) {
        int w = k * TPB + tid;
        if (w < NP * 8) {
            int p  = w >> 3;
            int d4 = w & 7;
            int pr = s_pair[p];
            int i  = pr >> 8;
            int j  = pr & 0xff;
            size_t a = ((size_t)j * VOC + s_adj[i]) * (ED / 4) + (size_t)d4;
            size_t c = ((size_t)i * VOC + s_adj[j]) * (ED / 4) + (size_t)d4;
            v4f prod = Wv[a] * Wv[c];
            // Output is never re-read: non-temporal store keeps L2 for the embedding hot set.
            __builtin_nontemporal_store(prod, outv + w);
        }
    }
}

extern "C" void kernel_launch(void* const* d_in, const int* in_sizes, int n_in,
                              void* d_out, int out_size, void* d_ws, size_t ws_size,
                              hipStream_t stream) {
    (void)in_sizes; (void)n_in; (void)out_size; (void)d_ws; (void)ws_size;
    const int*   x    = (const int*)d_in[0];
    const int*   offs = (const int*)d_in[1];
    const float* W    = (const float*)d_in[2];
    float*       out  = (float*)d_out;
    ffm_cross_kernel<<<NB, TPB, 0, stream>>>(x, offs, W, out);
}